// FCP_V1_2362232012979
// MI455X (gfx1250) — compile-verified
//
#include <hip/hip_runtime.h>
#include <math.h>
#include <stdint.h>

// ---------------- problem constants ----------------
#define B_     4
#define S_     2
#define C_     6
#define L0_    64000
#define NFFT_  512
#define HOP_   128
#define F_     257
#define T_     505          // frames: ((64000+512+512)-512)/128 + 1
#define TAPS_  20
#define LOUT_  65024        // (T_-1)*HOP_ + NFFT_
#define NSIG_  48           // B*S*C
#define NBFS_  2056         // B*F*S
#define LAMBDA_  2e-4f
#define FCP_EPS_ 1e-3f
#define YPAD_  544          // 19 leading zeros + 505 data + trailing zeros

typedef float v2f __attribute__((ext_vector_type(2)));
typedef float v8f __attribute__((ext_vector_type(8)));
typedef unsigned int u32x4 __attribute__((ext_vector_type(4)));
typedef int i32x4 __attribute__((ext_vector_type(4)));
typedef int i32x8 __attribute__((ext_vector_type(8)));

#define WMMA_F32(A,Bm,Cacc) \
  __builtin_amdgcn_wmma_f32_16x16x4_f32(false,(A),false,(Bm),(short)0,(Cacc),false,false)

#if defined(__has_builtin)
#if __has_builtin(__builtin_amdgcn_tensor_load_to_lds) && \
    __has_builtin(__builtin_amdgcn_s_wait_tensorcnt)
#define USE_TDM 1
#endif
#endif

#ifdef USE_TDM
// Pack a Tensor-DMA descriptor (D#) per CDNA5 ISA §8.3/8.4 and issue
// tensor_load_to_lds (clang-23 6-arg builtin: g0, g1, g2, g3, g4-pad, cpol).
// 2-D tile: tile0 elements per row, tile1 rows, row stride `stride0`
// (all in data_size units).  Groups 2/3 zero => <=2-D tensor.
__device__ __forceinline__ void tdm_load_2d(void* lds_dst, const void* gsrc,
                                            unsigned dsz_log2, unsigned tile0,
                                            unsigned tile1,
                                            unsigned long long stride0) {
  unsigned lds_off = (unsigned)(unsigned long long)(uintptr_t)lds_dst;
  unsigned long long ga = (unsigned long long)(uintptr_t)gsrc;
  u32x4 g0;
  g0.x = 1u;                                               // count=1, user D#
  g0.y = lds_off;                                          // lds_addr
  g0.z = (unsigned)(ga & 0xffffffffull);                   // global_addr lo
  g0.w = (unsigned)((ga >> 32) & 0x1ffffffull) | (2u << 30); // hi + type=2
  unsigned td0 = tile0;                                    // tensor_dim0
  unsigned td1 = tile1;                                    // tensor_dim1
  i32x8 g1;
  g1[0] = (int)(dsz_log2 << 16);                           // data_size
  g1[1] = (int)((td0 & 0xffffu) << 16);                    // tensor_dim0[15:0]
  g1[2] = (int)((td0 >> 16) | ((td1 & 0xffffu) << 16));    // td0 hi | td1 lo
  g1[3] = (int)((td1 >> 16) | ((tile0 & 0xffffu) << 16));  // td1 hi | tile_dim0
  g1[4] = (int)(tile1 & 0xffffu);                          // tile_dim1 (dim2=0)
  g1[5] = (int)(stride0 & 0xffffffffull);                  // dim0_stride lo
  g1[6] = (int)((stride0 >> 32) & 0xffffull);              // dim0_stride hi
  g1[7] = 0;
  i32x4 z4 = {0, 0, 0, 0};
  i32x8 z8 = {0, 0, 0, 0, 0, 0, 0, 0};
  __builtin_amdgcn_tensor_load_to_lds(g0, g1, z4, z4, z8, 0);
}
#endif

__device__ __forceinline__ float2 cmul(float2 a, float2 b) {
  return make_float2(a.x*b.x - a.y*b.y, a.x*b.y + a.y*b.x);
}
__device__ __forceinline__ float2 csub(float2 a, float2 b) {
  return make_float2(a.x - b.x, a.y - b.y);
}
__device__ __forceinline__ float sqrt_hann(int n) {
  return sqrtf(0.5f * (1.0f - cosf(6.28318530717958647692f * (float)n / (float)NFFT_)));
}

// ---------------- 1) STFT: direct DFT with LDS cos table ----------------
__global__ __launch_bounds__(256)
void k_stft(const float* __restrict__ ref, const float* __restrict__ mix,
            float2* __restrict__ SPH, float2* __restrict__ MXS) {
  int t   = blockIdx.x;          // frame
  int sig = blockIdx.y;          // 0..7 = ref(b,s), 8..31 = mix(b,c)
  int tid = threadIdx.x;
  const float* src;
  float2* dst;
  if (sig < 8) { src = ref + (size_t)sig * L0_;      dst = SPH + (size_t)sig * F_ * T_; }
  else         { src = mix + (size_t)(sig-8) * L0_;  dst = MXS + (size_t)(sig-8) * F_ * T_; }

  __shared__ float costab[NFFT_];
  __shared__ float xw[NFFT_];
  for (int i = tid; i < NFFT_; i += 256)
    costab[i] = cosf(6.28318530717958647692f * (float)i / (float)NFFT_);
  for (int n = tid; n < NFFT_; n += 256) {
    int g = t * HOP_ + n - (NFFT_/2);                 // center pad 256
    float x = (g >= 0 && g < L0_) ? src[g] : 0.0f;
    xw[n] = x * sqrt_hann(n);
  }
  __syncthreads();
  for (int f = tid; f < F_; f += 256) {
    float re = 0.f, im = 0.f;
    for (int n = 0; n < NFFT_; ++n) {
      int idx = (f * n) & (NFFT_-1);
      float v = xw[n];
      re += v * costab[idx];
      im -= v * costab[(idx + 384) & (NFFT_-1)];      // sin(th)=cos(th-pi/2)
    }
    dst[(size_t)f * T_ + t] = make_float2(re, im);
  }
}

// ---------------- 2) mean mic power + per-batch max ----------------
__global__ __launch_bounds__(256)
void k_power(const float2* __restrict__ MXS, float* __restrict__ POW,
             unsigned* __restrict__ PMAX) {
  int i = blockIdx.x * 256 + threadIdx.x;
  if (i >= B_ * F_ * T_) return;
  int b = i / (F_ * T_);
  int rem = i % (F_ * T_);
  int f = rem / T_, t = rem % T_;
  float s = 0.f;
  for (int c = 0; c < C_; ++c) {
    float2 v = MXS[((size_t)(b*C_ + c) * F_ + f) * T_ + t];
    s += v.x*v.x + v.y*v.y;
  }
  float p = s * (1.0f / (float)C_);
  POW[i] = p;
  atomicMax(&PMAX[b], __float_as_uint(p));   // p>=0: uint order == float order
}

__global__ __launch_bounds__(256)
void k_invpow(float* __restrict__ POW, const unsigned* __restrict__ PMAX) {
  int i = blockIdx.x * 256 + threadIdx.x;
  if (i >= B_ * F_ * T_) return;
  int b = i / (F_ * T_);
  float mx = __uint_as_float(PMAX[b]);
  POW[i] = 1.0f / (POW[i] + mx * FCP_EPS_ + 1e-6f);
}

// ---------------- 3) R|P via fp32 WMMA, one wave per (b,f,s) ----------------
// C[m,n] = sum_t A[t,m] * conj(Braw[t,n]);  A = inv_pow .* Toeplitz(sph),
// Braw cols 0..19 = Toeplitz(sph), 20..25 = mxs.  Branch-free inner loop:
// zero-padded LDS images make every fragment element an unconditional ds read.
// Rows m>=20 / cols n>=26 of the 32x32 tile are garbage-but-finite, never read.
__global__ __launch_bounds__(32)
void k_accum(const float2* __restrict__ SPH, const float2* __restrict__ MXS,
             const float* __restrict__ IPOW, float2* __restrict__ RP) {
  int blk = blockIdx.x;
  int b = blk / (F_ * S_);
  int f = (blk / S_) % F_;
  int s = blk % S_;
  int lane = threadIdx.x;
  int lm = lane & 15, lh = lane >> 4;

  __shared__ float2 sh_yp[YPAD_];            // [19 zeros][505][zeros]
  __shared__ float  sh_ip[512];              // zeros beyond 505 (kills K pad)
  __shared__ float2 sh_m[C_ * T_ + 8];       // flat mic rows + safety tail

  const float2* Yg = SPH + ((size_t)(b*S_ + s) * F_ + f) * T_;
  const float2* Mg = MXS + ((size_t)(b*C_) * F_ + f) * T_;
  const float*  Ig = IPOW + ((size_t)b * F_ + f) * T_;

#ifdef USE_TDM
  // DMA the three staging tiles; zero-fill only the pad regions (disjoint
  // from DMA-written bytes) while the TDM works.
  tdm_load_2d(&sh_yp[TAPS_-1], Yg, 3, T_, 1, T_);
  tdm_load_2d(sh_ip, Ig, 2, T_, 1, T_);
  tdm_load_2d(sh_m, Mg, 3, T_, C_, (unsigned long long)F_ * T_);
  {
    const float2 z2 = make_float2(0.f, 0.f);
    if (lane < TAPS_-1) sh_yp[lane] = z2;
    for (int i = TAPS_-1 + T_ + lane; i < YPAD_; i += 32) sh_yp[i] = z2;
    if (lane < 7) sh_ip[T_ + lane] = 0.f;
    if (lane < 8) sh_m[C_*T_ + lane] = z2;
  }
  __builtin_amdgcn_s_wait_tensorcnt(0);
#else
  {
    const float2 z2 = make_float2(0.f, 0.f);
    if (lane < TAPS_-1) sh_yp[lane] = z2;
    for (int i = TAPS_-1 + T_ + lane; i < YPAD_; i += 32) sh_yp[i] = z2;
    if (lane < 7) sh_ip[T_ + lane] = 0.f;
    if (lane < 8) sh_m[C_*T_ + lane] = z2;
    __builtin_prefetch(Yg, 0, 0);
    for (int i = lane; i < T_; i += 32) { sh_yp[TAPS_-1 + i] = Yg[i]; sh_ip[i] = Ig[i]; }
    for (int c = 0; c < C_; ++c)
      for (int i = lane; i < T_; i += 32) sh_m[c*T_ + i] = Mg[(size_t)c * F_ * T_ + i];
  }
#endif
  __syncthreads();

  const v8f Z8 = {0.f,0.f,0.f,0.f,0.f,0.f,0.f,0.f};
  v8f acc[2][2][4];
  #pragma unroll
  for (int mt = 0; mt < 2; ++mt)
    #pragma unroll
    for (int nt = 0; nt < 2; ++nt)
      #pragma unroll
      for (int q = 0; q < 4; ++q) acc[mt][nt][q] = Z8;

  // B-column source: n<20 -> Toeplitz, else mic (clamped; cols>=26 garbage)
  int ncol = lm;                               // per-lane column base (nt adds 16)

  for (int k0 = 0; k0 < 508; k0 += 4) {        // K = frame t, zero-padded
    int tb = k0 + (lh << 1);                   // A-side K pair base
    float ip0 = sh_ip[tb], ip1 = sh_ip[tb + 1];
    v2f ar[2], ai[2], bx[2], by[2];
    #pragma unroll
    for (int mt = 0; mt < 2; ++mt) {           // A: 16x4 (M=tap, K=t)
      int p = mt*16 + lm;
      float2 y0 = sh_yp[tb + p];               // Yt[t,p] = sh_yp[t+p]
      float2 y1 = sh_yp[tb + 1 + p];
      ar[mt][0] = ip0 * y0.x; ai[mt][0] = ip0 * y0.y;
      ar[mt][1] = ip1 * y1.x; ai[mt][1] = ip1 * y1.y;
    }
    #pragma unroll
    for (int nt = 0; nt < 2; ++nt) {           // B: 4x16 (K=t, N=col), raw
      int n = nt*16 + ncol;
      int cm = n - TAPS_; cm = cm < 0 ? 0 : (cm > C_-1 ? C_-1 : cm);
      #pragma unroll
      for (int v = 0; v < 2; ++v) {
        int t = k0 + (v << 1) + lh;
        float2 yv = sh_yp[t + n];              // Toeplitz candidate
        float2 mv = sh_m[cm * T_ + t];         // mic candidate
        float2 vv = (n < TAPS_) ? yv : mv;
        bx[nt][v] = vv.x; by[nt][v] = vv.y;
      }
    }
    #pragma unroll
    for (int mt = 0; mt < 2; ++mt)
      #pragma unroll
      for (int nt = 0; nt < 2; ++nt) {
        acc[mt][nt][0] = WMMA_F32(ar[mt], bx[nt], acc[mt][nt][0]); // Re: ar*x
        acc[mt][nt][1] = WMMA_F32(ai[mt], by[nt], acc[mt][nt][1]); // Re: ai*y
        acc[mt][nt][2] = WMMA_F32(ai[mt], bx[nt], acc[mt][nt][2]); // Im: ai*x
        acc[mt][nt][3] = WMMA_F32(ar[mt], by[nt], acc[mt][nt][3]); // Im: ar*y
      }
  }

  // A * conj(Braw): re = S(ar*x)+S(ai*y), im = S(ai*x)-S(ar*y)
  const float inv_t = 1.0f / (float)T_;
  float2* out = RP + (size_t)blk * 1024;       // 32x32 padded complex tile
  #pragma unroll
  for (int mt = 0; mt < 2; ++mt)
    #pragma unroll
    for (int nt = 0; nt < 2; ++nt)
      #pragma unroll
      for (int r = 0; r < 8; ++r) {
        int m = mt*16 + r + (lh ? 8 : 0);
        int n = nt*16 + lm;
        float cre = (acc[mt][nt][0][r] + acc[mt][nt][1][r]) * inv_t;
        float cim = (acc[mt][nt][2][r] - acc[mt][nt][3][r]) * inv_t;
        out[m*32 + n] = make_float2(cre, cim);
      }
}

// ---------------- 4) per-wave 20x20 complex solve, 6 RHS ----------------
__global__ __launch_bounds__(32)
void k_solve(const float2* __restrict__ RP, float2* __restrict__ GC,
             float* __restrict__ outG) {
  int blk = blockIdx.x;
  int lane = threadIdx.x;
  __shared__ float2 A[TAPS_][TAPS_ + C_];
  __shared__ float2 piv;
  const float2* in = RP + (size_t)blk * 1024;
  for (int idx = lane; idx < TAPS_*(TAPS_+C_); idx += 32) {
    int i = idx / (TAPS_+C_), j = idx % (TAPS_+C_);
    float2 v = in[i*32 + j];
    if (j == i) v.x += LAMBDA_;                    // diagonal loading
    A[i][j] = v;
  }
  __syncthreads();
  for (int k = 0; k < TAPS_; ++k) {
    if (lane == 0) {
      float2 d = A[k][k];
      float den = d.x*d.x + d.y*d.y + 1e-30f;
      piv = make_float2(d.x/den, -d.y/den);
    }
    __syncthreads();
    if (lane >= k && lane < TAPS_+C_) A[k][lane] = cmul(A[k][lane], piv);
    __syncthreads();
    for (int i = k+1; i < TAPS_; ++i) {
      float2 fkt = A[i][k];
      __syncthreads();
      if (lane >= k && lane < TAPS_+C_) A[i][lane] = csub(A[i][lane], cmul(fkt, A[k][lane]));
      __syncthreads();
    }
  }
  for (int k = TAPS_-1; k >= 1; --k) {
    for (int i = 0; i < k; ++i) {
      float2 fkt = A[i][k];
      __syncthreads();
      if (lane < C_) A[i][TAPS_+lane] = csub(A[i][TAPS_+lane], cmul(fkt, A[k][TAPS_+lane]));
      __syncthreads();
    }
  }
  for (int idx = lane; idx < TAPS_*C_; idx += 32) {
    int p = idx / C_, c = idx % C_;
    float2 g = A[p][TAPS_ + c];
    float2 gc = make_float2(g.x, -g.y);            // G.conj()
    GC[(size_t)blk * TAPS_*C_ + idx] = gc;
    outG[((size_t)blk * TAPS_*C_ + idx)*2 + 0] = gc.x;   // output #1
    outG[((size_t)blk * TAPS_*C_ + idx)*2 + 1] = gc.y;
  }
}

// ---------------- 5) filter apply via fp32 WMMA: spec = Yt * Gc ----------------
__global__ __launch_bounds__(32)
void k_apply(const float2* __restrict__ SPH, const float2* __restrict__ GC,
             float2* __restrict__ SPEC) {
  int blk = blockIdx.x;
  int b = blk / (F_ * S_);
  int f = (blk / S_) % F_;
  int s = blk % S_;
  int lane = threadIdx.x;
  int lm = lane & 15, lh = lane >> 4;
  __shared__ float2 sh_yp[YPAD_];
  __shared__ float2 sh_g[TAPS_][16];          // zero-padded N (cols 6..15 = 0)
  const float2* Yg = SPH + ((size_t)(b*S_ + s) * F_ + f) * T_;

#ifdef USE_TDM
  tdm_load_2d(&sh_yp[TAPS_-1], Yg, 3, T_, 1, T_);
  {
    const float2 z2 = make_float2(0.f, 0.f);
    if (lane < TAPS_-1) sh_yp[lane] = z2;
    for (int i = TAPS_-1 + T_ + lane; i < YPAD_; i += 32) sh_yp[i] = z2;
    for (int i = lane; i < TAPS_*16; i += 32) sh_g[i >> 4][i & 15] = z2;
  }
  __builtin_amdgcn_s_wait_tensorcnt(0);
  __syncthreads();
  for (int i = lane; i < TAPS_*C_; i += 32)
    sh_g[i / C_][i % C_] = GC[(size_t)blk*TAPS_*C_ + i];
#else
  {
    const float2 z2 = make_float2(0.f, 0.f);
    if (lane < TAPS_-1) sh_yp[lane] = z2;
    for (int i = TAPS_-1 + T_ + lane; i < YPAD_; i += 32) sh_yp[i] = z2;
    for (int i = lane; i < TAPS_*16; i += 32) sh_g[i >> 4][i & 15] = z2;
    __builtin_prefetch(Yg, 0, 0);
    for (int i = lane; i < T_; i += 32) sh_yp[TAPS_-1 + i] = Yg[i];
  }
  __syncthreads();
  for (int i = lane; i < TAPS_*C_; i += 32)
    sh_g[i / C_][i % C_] = GC[(size_t)blk*TAPS_*C_ + i];
#endif
  __syncthreads();

  const v8f Z8 = {0.f,0.f,0.f,0.f,0.f,0.f,0.f,0.f};
  for (int mt = 0; mt < 32; ++mt) {                // M = frames (505 -> 512)
    v8f arr = Z8, aii = Z8, ari = Z8, air = Z8;
    int t = mt*16 + lm;                            // A-row frame index
    #pragma unroll
    for (int k0 = 0; k0 < TAPS_; k0 += 4) {        // K = taps (exactly 20)
      v2f ar, ai, bx, by;
      #pragma unroll
      for (int kk = 0; kk < 2; ++kk) {
        int p = k0 + (lh << 1) + kk;               // p <= 19 always
        float2 y = sh_yp[t + p];                   // rows t>=505: garbage, masked later
        ar[kk] = y.x; ai[kk] = y.y;
      }
      #pragma unroll
      for (int v = 0; v < 2; ++v) {
        int p = k0 + (v << 1) + lh;
        float2 g = sh_g[p][lm];                    // cols >=6 are zero
        bx[v] = g.x; by[v] = g.y;
      }
      arr = WMMA_F32(ar, bx, arr);
      aii = WMMA_F32(ai, by, aii);
      ari = WMMA_F32(ar, by, ari);
      air = WMMA_F32(ai, bx, air);
    }
    #pragma unroll
    for (int r = 0; r < 8; ++r) {
      int tt = mt*16 + r + (lh ? 8 : 0);
      int c = lm;
      if (tt < T_ && c < C_) {
        float cre = arr[r] - aii[r];               // (a)(g): re = ar*x - ai*y
        float cim = ari[r] + air[r];
        SPEC[(((size_t)((b*S_ + s)*C_ + c)) * F_ + f) * T_ + tt] = make_float2(cre, cim);
      }
    }
  }
}

// ---------------- 6) overlap-add envelope ----------------
__global__ __launch_bounds__(256)
void k_env(float* __restrict__ ENV) {
  int t = blockIdx.x;
  for (int n = threadIdx.x; n < NFFT_; n += 256) {
    float w = sqrt_hann(n);
    atomicAdd(&ENV[t*HOP_ + n], w*w);
  }
}

// ---------------- 7) inverse DFT + windowed overlap-add ----------------
__global__ __launch_bounds__(256)
void k_istft(const float2* __restrict__ SPEC, float* __restrict__ YACC) {
  int t   = blockIdx.x;
  int sig = blockIdx.y;
  int tid = threadIdx.x;
  __shared__ float2 X[F_];
  __shared__ float costab[NFFT_];
  for (int i = tid; i < NFFT_; i += 256)
    costab[i] = cosf(6.28318530717958647692f * (float)i / (float)NFFT_);
  for (int f = tid; f < F_; f += 256)
    X[f] = SPEC[((size_t)sig * F_ + f) * T_ + t];
  __syncthreads();
  for (int n = tid; n < NFFT_; n += 256) {
    float acc = X[0].x + X[256].x * ((n & 1) ? -1.0f : 1.0f);
    for (int k = 1; k < 256; ++k) {
      int idx = (k * n) & (NFFT_-1);
      acc += 2.0f * (X[k].x * costab[idx] - X[k].y * costab[(idx + 384) & (NFFT_-1)]);
    }
    float x = acc * (1.0f / (float)NFFT_);
    atomicAdd(&YACC[(size_t)sig * LOUT_ + t*HOP_ + n], x * sqrt_hann(n));
  }
}

// ---------------- 8) normalize + crop -> audio output ----------------
__global__ __launch_bounds__(256)
void k_final(const float* __restrict__ YACC, const float* __restrict__ ENV,
             float* __restrict__ outA) {
  int i = blockIdx.x * 256 + threadIdx.x;
  if (i >= NSIG_ * L0_) return;
  int sig = i / L0_;
  int n   = i % L0_;
  int gi  = n + (NFFT_/2);                         // crop center pad
  float e = ENV[gi];
  e = (e > 1e-11f) ? e : 1.0f;
  outA[i] = YACC[(size_t)sig * LOUT_ + gi] / e;
}

// ---------------- host launcher ----------------
extern "C" void kernel_launch(void* const* d_in, const int* in_sizes, int n_in,
                              void* d_out, int out_size, void* d_ws, size_t ws_size,
                              hipStream_t stream) {
  const float* ref = (const float*)d_in[0];   // [4,2,64000]
  const float* mix = (const float*)d_in[1];   // [4,6,64000]
  float* out = (float*)d_out;                 // [G.conj interleaved c64][audio]

  char* ws = (char*)d_ws;
  size_t off = 0;
  auto alloc = [&](size_t bytes) -> char* {
    char* p = ws + off; off += (bytes + 255) & ~(size_t)255; return p;
  };
  float2*   SPH  = (float2*)  alloc((size_t)(B_*S_) * F_ * T_ * sizeof(float2));
  float2*   MXS  = (float2*)  alloc((size_t)(B_*C_) * F_ * T_ * sizeof(float2));
  float*    IPOW = (float*)   alloc((size_t)B_ * F_ * T_ * sizeof(float));
  unsigned* PMAX = (unsigned*)alloc(B_ * sizeof(unsigned));
  float2*   RP   = (float2*)  alloc((size_t)NBFS_ * 1024 * sizeof(float2));
  float2*   GC   = (float2*)  alloc((size_t)NBFS_ * TAPS_ * C_ * sizeof(float2));
  float2*   SPEC = (float2*)  alloc((size_t)NSIG_ * F_ * T_ * sizeof(float2));
  float*    YACC = (float*)   alloc((size_t)NSIG_ * LOUT_ * sizeof(float));
  float*    ENV  = (float*)   alloc((size_t)LOUT_ * sizeof(float));

  (void)hipMemsetAsync(PMAX, 0, B_ * sizeof(unsigned), stream);
  (void)hipMemsetAsync(YACC, 0, (size_t)NSIG_ * LOUT_ * sizeof(float), stream);
  (void)hipMemsetAsync(ENV,  0, (size_t)LOUT_ * sizeof(float), stream);

  k_stft<<<dim3(T_, 32), 256, 0, stream>>>(ref, mix, SPH, MXS);
  int bft = B_ * F_ * T_;
  k_power <<<(bft + 255)/256, 256, 0, stream>>>(MXS, IPOW, PMAX);
  k_invpow<<<(bft + 255)/256, 256, 0, stream>>>(IPOW, PMAX);
  k_accum<<<NBFS_, 32, 0, stream>>>(SPH, MXS, IPOW, RP);
  k_solve<<<NBFS_, 32, 0, stream>>>(RP, GC, out);
  k_apply<<<NBFS_, 32, 0, stream>>>(SPH, GC, SPEC);
  k_env<<<T_, 256, 0, stream>>>(ENV);
  k_istft<<<dim3(T_, NSIG_), 256, 0, stream>>>(SPEC, YACC);
  int na = NSIG_ * L0_;
  k_final<<<(na + 255)/256, 256, 0, stream>>>(YACC, ENV,
      out + (size_t)NBFS_ * TAPS_ * C_ * 2);   // audio after G block
}